// DGCNN_54700703482024
// MI455X (gfx1250) — compile-verified
//
#include <hip/hip_runtime.h>

typedef __attribute__((ext_vector_type(2))) float v2f;
typedef __attribute__((ext_vector_type(8))) float v8f;

#define NB   16
#define NP   2048
#define ND   64
#define NK   16
#define NOUT 128
#define NC   256   // fused GEMM output channels: [U | V]

// ---------------------------------------------------------------------------
// Helpers
// ---------------------------------------------------------------------------

// 32-bit LDS byte offset of a shared-memory object (flat-address low bits are
// the wave-relative LDS address on CDNA5).
__device__ __forceinline__ unsigned lds_offset_of(const void* p) {
  return (unsigned)(unsigned long long)p;
}

// Async-copy one 16x64 f32 tile (4 KB) global -> LDS using 8 per-lane
// GLOBAL_LOAD_ASYNC_TO_LDS_B128 ops (tracked by ASYNCcnt, no VGPR bounce).
__device__ __forceinline__ void async_tile_4k(const float* __restrict__ g,
                                              const float* l, int lane) {
  const unsigned base = lds_offset_of(l);
  const float4* g4 = (const float4*)g;
#pragma unroll
  for (int i = 0; i < 8; ++i) {
    const unsigned lo = base + (unsigned)(lane + i * 32) * 16u;
    const float4* gp = g4 + (lane + i * 32);
    asm volatile("global_load_async_to_lds_b128 %0, %1, off"
                 :: "v"(lo), "v"(gp)
                 : "memory");
  }
}

// Insert candidate (cv, ci) into ascending-sorted top-16 kept in registers.
__device__ __forceinline__ void topk_insert(float (&bd)[NK], int (&bi)[NK],
                                            float cv, int ci) {
  if (cv < bd[NK - 1]) {
    bd[NK - 1] = cv;
    bi[NK - 1] = ci;
#pragma unroll
    for (int j = NK - 1; j > 0; --j) {
      if (bd[j] < bd[j - 1]) {
        float td = bd[j]; bd[j] = bd[j - 1]; bd[j - 1] = td;
        int   ti = bi[j]; bi[j] = bi[j - 1]; bi[j - 1] = ti;
      }
    }
  }
}

// ---------------------------------------------------------------------------
// Wc[o][d]      = W[o][d] - W[o][64+d]   (o in [0,128))
// Wc[128+o][d]  = W[o][64+d]
// ---------------------------------------------------------------------------
__global__ __launch_bounds__(256) void wc_kernel(const float* __restrict__ W,
                                                 float* __restrict__ Wc) {
  const int i = blockIdx.x * blockDim.x + threadIdx.x;
  if (i < NOUT * ND) {
    const int o = i >> 6, d = i & 63;
    const float w1 = W[o * (2 * ND) + d];
    const float w2 = W[o * (2 * ND) + ND + d];
    Wc[o * ND + d]          = w1 - w2;
    Wc[(NOUT + o) * ND + d] = w2;
  }
}

// ---------------------------------------------------------------------------
// sq[p] = ||x[p]||^2  (hoisted out of the KNN tile loop)
// ---------------------------------------------------------------------------
__global__ __launch_bounds__(256) void sq_kernel(const float* __restrict__ x,
                                                 float* __restrict__ sq) {
  const int p = blockIdx.x * blockDim.x + threadIdx.x;
  if (p < NB * NP) {
    const float* xp = x + (size_t)p * ND;
    float s = 0.f;
#pragma unroll
    for (int d = 0; d < ND; ++d) s += xp[d] * xp[d];
    sq[p] = s;
  }
}

// ---------------------------------------------------------------------------
// KNN: one workgroup = 16 rows x all 2048 cols of one cloud's distance matrix.
// Dot tiles via V_WMMA_F32_16X16X4_F32; column tiles double-buffered with
// GLOBAL_LOAD_ASYNC_TO_LDS_B128 prefetch; rank key = sq[j] - 2*dot.
// ---------------------------------------------------------------------------
__global__ __launch_bounds__(256) void knn_kernel(const float* __restrict__ x,
                                                  const float* __restrict__ sq,
                                                  int* __restrict__ nn_out) {
  const int b = blockIdx.y;
  const int rowBase = blockIdx.x << 4;
  const float* __restrict__ xb  = x  + (size_t)b * NP * ND;
  const float* __restrict__ sqg = sq + (size_t)b * NP;

  __shared__ alignas(16) float As[16][ND];          // 4 KB row block
  __shared__ alignas(16) float Bs[8][2][16][ND];    // 64 KB double-buffered tiles
  __shared__ float sqB[8][16];
  __shared__ float tileS[8][16][17];                // padded dot tiles

  const int tid  = threadIdx.x;
  const int wave = tid >> 5;
  const int lane = tid & 31;
  const int r    = lane & 15;
  const int hi   = lane >> 4;

  // Prefetch first column tile for this wave (ASYNCcnt path).
  async_tile_4k(xb + (size_t)(wave << 4) * ND, &Bs[wave][0][0][0], lane);

  for (int i = tid; i < 16 * ND; i += 256)
    As[i >> 6][i & 63] = xb[(size_t)(rowBase + (i >> 6)) * ND + (i & 63)];
  __syncthreads();

  float bd[NK]; int bi[NK];
#pragma unroll
  for (int k = 0; k < NK; ++k) { bd[k] = 3.0e38f; bi[k] = 0; }

  for (int s = 0; s < 16; ++s) {
    const int cur = s & 1;
    const int colBase = (wave + (s << 3)) << 4;

    // Prefetch next tile into the other buffer, then wait for current tile.
    if (s + 1 < 16) {
      const int nxtBase = (wave + ((s + 1) << 3)) << 4;
      async_tile_4k(xb + (size_t)nxtBase * ND, &Bs[wave][cur ^ 1][0][0], lane);
      asm volatile("s_wait_asynccnt 0x8" ::: "memory");
    } else {
      asm volatile("s_wait_asynccnt 0x0" ::: "memory");
    }

    if (lane < 16) sqB[wave][lane] = sqg[colBase + lane];

    // 16x16 dot tile: 16 WMMA f32 16x16x4 steps over D=64.
    v8f c = {};
#pragma unroll
    for (int kk = 0; kk < ND; kk += 4) {
      v2f a, bb;
      a.x  = As[r][kk + 2 * hi];
      a.y  = As[r][kk + 2 * hi + 1];
      bb.x = Bs[wave][cur][r][kk + 2 * hi];
      bb.y = Bs[wave][cur][r][kk + 2 * hi + 1];
      c = __builtin_amdgcn_wmma_f32_16x16x4_f32(false, a, false, bb, (short)0, c,
                                                false, false);
    }

#pragma unroll
    for (int v = 0; v < 8; ++v) tileS[wave][v + 8 * hi][r] = c[v];
    __syncthreads();

    if (lane < 16) {
#pragma unroll
      for (int n = 0; n < 16; ++n) {
        const int col = colBase + n;
        const float cv = sqB[wave][n] - 2.0f * tileS[wave][r][n];
        if (col != rowBase + r)  // exclude self (reference adds 1e10 on diag)
          topk_insert(bd, bi, cv, col);
      }
    }
    __syncthreads();
  }

  // Merge the 8 per-wave top-16 lists per row; alias candidate store over Bs.
  float* candD = &Bs[0][0][0][0];            // 2048 floats (8 KB)
  int*   candI = (int*)(&Bs[2][0][0][0]);    // 2048 ints   (8 KB)
  if (lane < 16) {
#pragma unroll
    for (int k = 0; k < NK; ++k) {
      candD[(wave * 16 + r) * NK + k] = bd[k];
      candI[(wave * 16 + r) * NK + k] = bi[k];
    }
  }
  __syncthreads();

  if (tid < 16) {
    float fd[NK]; int fi[NK];
#pragma unroll
    for (int k = 0; k < NK; ++k) { fd[k] = 3.0e38f; fi[k] = 0; }
    for (int w = 0; w < 8; ++w)
#pragma unroll
      for (int k = 0; k < NK; ++k)
        topk_insert(fd, fi, candD[(w * 16 + tid) * NK + k],
                    candI[(w * 16 + tid) * NK + k]);
    const size_t row = (size_t)b * NP + rowBase + tid;
#pragma unroll
    for (int k = 0; k < NK; ++k) nn_out[row * NK + k] = fi[k];
  }
}

// ---------------------------------------------------------------------------
// Y[b] = Xb @ Wc^T : [2048x64]x[64x256] per batch, WMMA f32 16x16x4.
// Both Wc tiles per wave async-loaded up front (wave-private LDS, no barriers
// in the compute loop).
// ---------------------------------------------------------------------------
__global__ __launch_bounds__(256) void edge_gemm_kernel(const float* __restrict__ x,
                                                        const float* __restrict__ Wc,
                                                        float* __restrict__ Y) {
  const int b = blockIdx.y;
  const int rowBase = blockIdx.x << 4;
  const float* __restrict__ xb = x + (size_t)b * NP * ND;

  __shared__ alignas(16) float As[16][ND];        // 4 KB
  __shared__ alignas(16) float Ws[8][2][16][ND];  // 64 KB Wc tiles

  const int tid  = threadIdx.x;
  const int wave = tid >> 5;
  const int lane = tid & 31;
  const int r    = lane & 15;
  const int hi   = lane >> 4;

  // Kick off both Wc tile copies for this wave (ASYNCcnt path).
  async_tile_4k(Wc + (size_t)(((wave << 1) + 0) << 4) * ND, &Ws[wave][0][0][0], lane);
  async_tile_4k(Wc + (size_t)(((wave << 1) + 1) << 4) * ND, &Ws[wave][1][0][0], lane);

  for (int i = tid; i < 16 * ND; i += 256)
    As[i >> 6][i & 63] = xb[(size_t)(rowBase + (i >> 6)) * ND + (i & 63)];
  __syncthreads();

#pragma unroll
  for (int t = 0; t < 2; ++t) {
    if (t == 0) asm volatile("s_wait_asynccnt 0x8" ::: "memory");
    else        asm volatile("s_wait_asynccnt 0x0" ::: "memory");

    const int colBase = ((wave << 1) + t) << 4;  // 16 output channels

    v8f c = {};
#pragma unroll
    for (int kk = 0; kk < ND; kk += 4) {
      v2f a, w;
      a.x = As[r][kk + 2 * hi];
      a.y = As[r][kk + 2 * hi + 1];
      w.x = Ws[wave][t][r][kk + 2 * hi];
      w.y = Ws[wave][t][r][kk + 2 * hi + 1];
      c = __builtin_amdgcn_wmma_f32_16x16x4_f32(false, a, false, w, (short)0, c,
                                                false, false);
    }

#pragma unroll
    for (int v = 0; v < 8; ++v) {
      const int M = v + 8 * hi;
      Y[((size_t)b * NP + rowBase + M) * NC + colBase + r] = c[v];
    }
  }
}

// ---------------------------------------------------------------------------
// out[p,o] = relu( U[p,o] + max_k V[nn[p,k],o] + bias[o] )
// ---------------------------------------------------------------------------
__global__ __launch_bounds__(128) void finalize_kernel(const float* __restrict__ Y,
                                                       const int* __restrict__ nn,
                                                       const float* __restrict__ bias,
                                                       float* __restrict__ out) {
  const int p = blockIdx.x;   // global point id
  const int o = threadIdx.x;  // channel
  const int b = p >> 11;      // p / 2048
  __shared__ int nbr[NK];
  if (o < NK) nbr[o] = nn[(size_t)p * NK + o];
  __syncthreads();

  const float u = Y[(size_t)p * NC + o];
  float m = -3.0e38f;
#pragma unroll
  for (int k = 0; k < NK; ++k) {
    const int j = nbr[k];  // within-batch index
    m = fmaxf(m, Y[(((size_t)b << 11) + j) * NC + NOUT + o]);
  }
  const float h = u + m + bias[o];
  out[(size_t)p * NOUT + o] = fmaxf(h, 0.0f);
}

// ---------------------------------------------------------------------------
extern "C" void kernel_launch(void* const* d_in, const int* in_sizes, int n_in,
                              void* d_out, int out_size, void* d_ws, size_t ws_size,
                              hipStream_t stream) {
  const float* x    = (const float*)d_in[0];   // [32768, 64]
  // d_in[1] = pos (unused), d_in[2] = batch (unused: uniform segments)
  const float* W    = (const float*)d_in[3];   // [128, 128]
  const float* bias = (const float*)d_in[4];   // [128]
  float* out = (float*)d_out;                  // [32768, 128]

  // Workspace layout: Wc (64KB) | nn (2MB) | sq (128KB) | Y (32MB)
  char* ws = (char*)d_ws;
  float* Wc = (float*)ws;
  int*   nn = (int*)(ws + (64u << 10));
  float* sq = (float*)(ws + (64u << 10) + (size_t)NB * NP * NK * sizeof(int));
  float* Y  = (float*)(ws + (64u << 10) + (size_t)NB * NP * NK * sizeof(int) +
                       (size_t)NB * NP * sizeof(float));

  wc_kernel<<<(NOUT * ND + 255) / 256, 256, 0, stream>>>(W, Wc);
  sq_kernel<<<(NB * NP + 255) / 256, 256, 0, stream>>>(x, sq);
  knn_kernel<<<dim3(NP / 16, NB), 256, 0, stream>>>(x, sq, nn);
  edge_gemm_kernel<<<dim3(NP / 16, NB), 256, 0, stream>>>(x, Wc, Y);
  finalize_kernel<<<NB * NP, NOUT, 0, stream>>>(Y, nn, bias, out);
}